// FlowLayer_28724741276121
// MI455X (gfx1250) — compile-verified
//
#include <hip/hip_runtime.h>
#include <hip/hip_bf16.h>

#define AS1 __attribute__((address_space(1)))
#define AS3 __attribute__((address_space(3)))

typedef int v4i __attribute__((vector_size(16)));

#ifndef __has_builtin
#define __has_builtin(x) 0
#endif

#if __has_builtin(__builtin_amdgcn_global_load_async_to_lds_b128)
#define HAVE_ASYNC_B128 1
#else
#define HAVE_ASYNC_B128 0
#endif

static constexpr int C_CH  = 16;
static constexpr int ROWF  = 48;    // C*D floats per node row
static constexpr int SLOTF = 100;   // padded per-thread slot (dwords): gcd(100,64)=4 -> 2-way LDS conflicts
static constexpr int EBLK  = 128;   // edge-kernel block size -> 50 KB LDS/block

// ---- async gather of one 192B node row into LDS -----------------------------
// One base global addr + one base LDS addr; the 16B step rides in the
// instruction immediate (HW adds INST_OFFSET to BOTH dsaddr and memaddr).
__device__ __forceinline__ void async_row48(const float* __restrict__ g, float* l) {
#if HAVE_ASYNC_B128
  AS1 v4i* gp = (AS1 v4i*)g;
  AS3 v4i* lp = (AS3 v4i*)l;
#define AR48(i) __builtin_amdgcn_global_load_async_to_lds_b128(gp, lp, 16 * (i), 0)
  AR48(0);  AR48(1);  AR48(2);  AR48(3);
  AR48(4);  AR48(5);  AR48(6);  AR48(7);
  AR48(8);  AR48(9);  AR48(10); AR48(11);
#undef AR48
#else
  unsigned           lo = (unsigned)(__SIZE_TYPE__)(AS3 char*)l;  // 32-bit LDS offset
  unsigned long long ga = (unsigned long long)g;
#pragma unroll
  for (int i = 0; i < 12; ++i)
    asm volatile("global_load_async_to_lds_b128 %0, %1, off"
                 :: "v"(lo + 16u * i), "v"(ga + 16ull * i) : "memory");
#endif
}

__device__ __forceinline__ void wait_async0() {
#if __has_builtin(__builtin_amdgcn_s_wait_asynccnt)
  __builtin_amdgcn_s_wait_asynccnt(0);
  asm volatile("" ::: "memory");
#else
  asm volatile("s_wait_asynccnt 0" ::: "memory");
#endif
}

// ---- per-edge: weighted sphere log maps, scatter-added by sender -----------
// Defined FIRST so the disasm snippet shows this (hot) kernel.
__global__ __launch_bounds__(EBLK) void edge_kernel(
    const float* __restrict__ x, const float* __restrict__ w,
    const int* __restrict__ snd, const int* __restrict__ rcv,
    float* __restrict__ agg, int E) {
  __shared__ float smem[EBLK * SLOTF];
  int e = blockIdx.x * EBLK + threadIdx.x;
  float* slot = &smem[threadIdx.x * SLOTF];
  if (e >= E) return;

  int s = snd[e], r = rcv[e];
  async_row48(x + (size_t)s * ROWF, slot);          // p row -> LDS (async DMA)
  async_row48(x + (size_t)r * ROWF, slot + ROWF);   // q row -> LDS (async DMA)
  float ew = w[e];                                  // overlaps the async gather
  wait_async0();

  float* ag = agg + (size_t)s * ROWF;
#pragma unroll
  for (int c = 0; c < C_CH; ++c) {
    float px = slot[c * 3 + 0], py = slot[c * 3 + 1], pz = slot[c * 3 + 2];
    float qx = slot[ROWF + c * 3 + 0], qy = slot[ROWF + c * 3 + 1], qz = slot[ROWF + c * 3 + 2];
    float cs = px * qx + py * qy + pz * qz;
    cs = fminf(1.0f, fmaxf(-1.0f, cs));
    float ux = qx - cs * px, uy = qy - cs * py, uz = qz - cs * pz;
    float un = sqrtf(ux * ux + uy * uy + uz * uz);
    float f  = ew * acosf(cs) / (un + 1e-12f);
    atomicAdd(&ag[c * 3 + 0], f * ux);
    atomicAdd(&ag[c * 3 + 1], f * uy);
    atomicAdd(&ag[c * 3 + 2], f * uz);
  }
}

// ---- deg[n] = segment_sum(edge_w, senders) ---------------------------------
__global__ void deg_kernel(const float* __restrict__ w, const int* __restrict__ snd,
                           float* __restrict__ deg, int E) {
  int e = blockIdx.x * blockDim.x + threadIdx.x;
  if (e < E) atomicAdd(&deg[snd[e]], w[e]);
}

// ---- per (node,channel): step-size logic + sphere exp map ------------------
__global__ void node_kernel(const float* __restrict__ xin, const float* __restrict__ agg,
                            const float* __restrict__ deg, const float* __restrict__ t_sqrt,
                            const float* __restrict__ delta_sqrt,
                            float* __restrict__ xout, int NC) {
  int i = blockIdx.x * blockDim.x + threadIdx.x;  // i = n*16 + c
  if (i >= NC) return;
  int c = i & 15, n = i >> 4;

  float dg = deg[n] + 1e-12f;
  const float* a = agg + (size_t)i * 3;
  float vx = -a[0] / dg, vy = -a[1] / dg, vz = -a[2] / dg;   // laplace(x)

  float nrm = sqrtf(vx * vx + vy * vy + vz * vz + 2.2204460492503131e-16f); // +EPS64
  float dl  = delta_sqrt[c]; dl *= dl;
  float alp = 1.0f / (1.0f + expf(-(nrm - dl)));
  float scale = (nrm * alp <= 1.0f) ? alp : (1.0f / nrm);     // MAX_STEP = 1
  float tc = t_sqrt[c]; tc = tc * tc * 0.5f;                  // t / N_STEPS
  float m = -scale * tc;
  vx *= m; vy *= m; vz *= m;

  const float* p = xin + (size_t)i * 3;
  float px = p[0], py = p[1], pz = p[2];
  float nn = sqrtf(vx * vx + vy * vy + vz * vz);
  float cn = cosf(nn);
  float sc = (nn > 1e-20f) ? (sinf(nn) / nn) : 1.0f;          // sinc(nn/pi)
  float yx = cn * px + sc * vx;
  float yy = cn * py + sc * vy;
  float yz = cn * pz + sc * vz;
  float inv = 1.0f / sqrtf(yx * yx + yy * yy + yz * yz);
  float* o = xout + (size_t)i * 3;
  o[0] = yx * inv; o[1] = yy * inv; o[2] = yz * inv;
}

extern "C" void kernel_launch(void* const* d_in, const int* in_sizes, int n_in,
                              void* d_out, int out_size, void* d_ws, size_t ws_size,
                              hipStream_t stream) {
  const float* nodes      = (const float*)d_in[0];
  const float* edge_w     = (const float*)d_in[1];
  const float* t_sqrt     = (const float*)d_in[2];
  const float* delta_sqrt = (const float*)d_in[3];
  const int*   senders    = (const int*)d_in[4];
  const int*   receivers  = (const int*)d_in[5];
  float*       out        = (float*)d_out;

  const int N = in_sizes[0] / ROWF;
  const int E = in_sizes[1];

  float* deg  = (float*)d_ws;                 // N
  float* agg  = deg + N;                      // N*48
  float* xtmp = agg + (size_t)N * ROWF;       // N*48

  (void)hipMemsetAsync(deg, 0, (size_t)N * sizeof(float), stream);
  deg_kernel<<<(E + 255) / 256, 256, 0, stream>>>(edge_w, senders, deg, E);

  const float* xin = nodes;
  float* xouts[2] = {xtmp, out};
  for (int s = 0; s < 2; ++s) {
    (void)hipMemsetAsync(agg, 0, (size_t)N * ROWF * sizeof(float), stream);
    edge_kernel<<<(E + EBLK - 1) / EBLK, EBLK, 0, stream>>>(xin, edge_w, senders,
                                                            receivers, agg, E);
    node_kernel<<<(N * C_CH + 255) / 256, 256, 0, stream>>>(xin, agg, deg, t_sqrt,
                                                            delta_sqrt, xouts[s], N * C_CH);
    xin = xouts[s];
  }
}